// PhysicsGuidedLoss_74749610819802
// MI455X (gfx1250) — compile-verified
//
#include <hip/hip_runtime.h>

#define LAMBDA_PHY 0.3f

typedef __attribute__((ext_vector_type(2))) float v2f;
typedef __attribute__((ext_vector_type(8))) float v8f;

// ---------------------------------------------------------------------------
// Full 32-lane sum via V_WMMA_F32_16X16X4_F32.
// A (16x4 f32, 2 VGPRs): lanes 0-15 hold {K0,K1}, lanes 16-31 hold {K2,K3}.
// We place the per-lane partial in K0/K2 and zero in K1/K3, B = ones(4x16):
//   D[m][n] = s[m] + s[m+16]  (replicated across n)
// C/D layout: lane n<16 holds rows 0..7 in VGPR 0..7, lane n>=16 rows 8..15.
// Sum the 8 D registers per lane, then one shfl_xor(16) -> total in all lanes.
// Requires EXEC == all ones (call only from reconverged full waves).
// ---------------------------------------------------------------------------
__device__ __forceinline__ float wave_sum32(float s) {
    v2f a; a[0] = s;    a[1] = 0.0f;
    v2f b; b[0] = 1.0f; b[1] = 1.0f;
    v8f c; c[0]=0.f; c[1]=0.f; c[2]=0.f; c[3]=0.f; c[4]=0.f; c[5]=0.f; c[6]=0.f; c[7]=0.f;
    v8f d = __builtin_amdgcn_wmma_f32_16x16x4_f32(
        /*neg_a=*/false, a, /*neg_b=*/false, b,
        /*c_mod=*/(short)0, c, /*reuse_a=*/false, /*reuse_b=*/false);
    float v = ((d[0] + d[1]) + (d[2] + d[3])) + ((d[4] + d[5]) + (d[6] + d[7]));
    v += __shfl_xor(v, 16, 32);
    return v;  // every lane holds the full 32-lane sum
}

// Block (256 threads = 8 wave32) sum. Result valid on thread 0.
__device__ __forceinline__ float block_sum256(float s, float* lds) {
    const int lane = threadIdx.x & 31;
    const int wid  = threadIdx.x >> 5;
    float w = wave_sum32(s);                 // all 8 waves fully converged here
    if (lane == 0) lds[wid] = w;
    __syncthreads();
    float r = 0.0f;
    if (wid == 0) {                          // wave-uniform branch: EXEC all-1 inside
        float x = (threadIdx.x < 8) ? lds[threadIdx.x & 7] : 0.0f;
        r = wave_sum32(x);
    }
    return r;
}

// ---------------------------------------------------------------------------
// Kernel 1: dense MSE partial sums over B*N floats (float4 vectorized).
// ---------------------------------------------------------------------------
__global__ void data_loss_partial(const float4* __restrict__ pred4,
                                  const float4* __restrict__ targ4,
                                  const float*  __restrict__ pred,
                                  const float*  __restrict__ targ,
                                  float* __restrict__ partials,
                                  int n4, int ntail) {
    __shared__ float lds[8];
    float acc = 0.0f;
    const int stride = gridDim.x * blockDim.x;
    for (int i = blockIdx.x * blockDim.x + threadIdx.x; i < n4; i += stride) {
        float4 p = pred4[i];
        float4 t = targ4[i];
        float dx = p.x - t.x, dy = p.y - t.y, dz = p.z - t.z, dw = p.w - t.w;
        acc = fmaf(dx, dx, acc);
        acc = fmaf(dy, dy, acc);
        acc = fmaf(dz, dz, acc);
        acc = fmaf(dw, dw, acc);
    }
    if (blockIdx.x == 0 && threadIdx.x == 0) {   // scalar tail (0..3 elems)
        for (int i = 4 * n4; i < 4 * n4 + ntail; ++i) {
            float d = pred[i] - targ[i];
            acc = fmaf(d, d, acc);
        }
    }
    float r = block_sum256(acc, lds);            // reconverged before WMMA
    if (threadIdx.x == 0) partials[blockIdx.x] = r;
}

// ---------------------------------------------------------------------------
// Kernel 2: physics residual partial sums. One thread per edge, loop over B.
// Edge metadata (src, dst, c0..c2) loaded once; 4 L2-resident gathers per b.
// ---------------------------------------------------------------------------
__global__ void phy_loss_partial(const float* __restrict__ pred,
                                 const float* __restrict__ prev,
                                 const float* __restrict__ c0,
                                 const float* __restrict__ c1,
                                 const float* __restrict__ c2,
                                 const long long* __restrict__ eidx,
                                 float* __restrict__ partials,
                                 int E, int N, int Bn) {
    __shared__ float lds[8];
    const int e = blockIdx.x * blockDim.x + threadIdx.x;
    float acc = 0.0f;
    if (e < E) {
        const int src = (int)eidx[e];
        const int dst = (int)eidx[E + e];
        const float k0 = c0[e];
        const float k1 = c1[e];
        const float k2 = c2[e];
        const float* __restrict__ ps = pred + src;
        const float* __restrict__ pd = pred + dst;
        const float* __restrict__ vs = prev + src;
        const float* __restrict__ vd = prev + dst;
        int off = 0;
#pragma unroll 4
        for (int b = 0; b < Bn; ++b) {
            float e0 = fmaf(k0, ps[off], fmaf(k1, vs[off], k2 * vd[off]));
            float r  = pd[off] - e0;
            acc = fmaf(r, r, acc);
            off += N;
        }
    }
    float r = block_sum256(acc, lds);            // reconverged before WMMA
    if (threadIdx.x == 0) partials[blockIdx.x] = r;
}

// ---------------------------------------------------------------------------
// Kernel 3: single-wave finalize. Deterministic ordered accumulation of block
// partials, WMMA cross-lane reduce, write (total, data_loss, phy_loss).
// ---------------------------------------------------------------------------
__global__ void finalize_losses(const float* __restrict__ dpart, int nd,
                                const float* __restrict__ ppart, int np,
                                float* __restrict__ out,
                                float invBN, float invBE) {
    const int lane = threadIdx.x;   // 32 threads, one wave
    float sd = 0.0f, sp = 0.0f;
    for (int i = lane; i < nd; i += 32) sd += dpart[i];
    for (int i = lane; i < np; i += 32) sp += ppart[i];
    float dsum = wave_sum32(sd);    // loops reconverged; EXEC all-1
    float psum = wave_sum32(sp);
    if (lane == 0) {
        float data_loss = dsum * invBN;
        float phy_loss  = psum * invBE;
        out[0] = fmaf(LAMBDA_PHY, phy_loss, data_loss);
        out[1] = data_loss;
        out[2] = phy_loss;
    }
}

extern "C" void kernel_launch(void* const* d_in, const int* in_sizes, int n_in,
                              void* d_out, int out_size, void* d_ws, size_t ws_size,
                              hipStream_t stream) {
    const float*     pred = (const float*)d_in[0];
    const float*     targ = (const float*)d_in[1];
    const float*     prev = (const float*)d_in[2];
    const float*     c0   = (const float*)d_in[3];
    const float*     c1   = (const float*)d_in[4];
    const float*     c2   = (const float*)d_in[5];
    const long long* eidx = (const long long*)d_in[6];

    const int BN = in_sizes[0];      // B*N = 640000
    const int E  = in_sizes[3];      // 320000
    const int Bn = 64;
    const int N  = BN / Bn;

    float* out = (float*)d_out;
    float* ws  = (float*)d_ws;

    const int TB = 256;
    const int n4    = BN >> 2;
    const int ntail = BN & 3;
    int nb_a = 512;
    if (nb_a * TB > n4) nb_a = (n4 + TB - 1) / TB;
    if (nb_a < 1) nb_a = 1;
    const int nb_b = (E + TB - 1) / TB;          // 1250

    float* dpart = ws;
    float* ppart = ws + nb_a;

    data_loss_partial<<<nb_a, TB, 0, stream>>>((const float4*)pred,
                                               (const float4*)targ,
                                               pred, targ, dpart, n4, ntail);
    phy_loss_partial<<<nb_b, TB, 0, stream>>>(pred, prev, c0, c1, c2, eidx,
                                              ppart, E, N, Bn);
    finalize_losses<<<1, 32, 0, stream>>>(dpart, nb_a, ppart, nb_b, out,
                                          1.0f / (float)BN,
                                          1.0f / ((float)Bn * (float)E));
}